// GCNLSTMInsiderThreat_21474836480577
// MI455X (gfx1250) — compile-verified
//
#include <hip/hip_runtime.h>
#include <math.h>

// ---------------------------------------------------------------------------
// Problem constants (match reference)
// ---------------------------------------------------------------------------
#define NUM_GRAPHS 512
#define N_NODES    131072
#define N_EDGES    4194304
#define F_IN       54
#define H_GCN      16
#define H_LSTM     32
#define N_CLASSES  192
#define SEQ_T      1024

typedef __attribute__((ext_vector_type(2))) float v2f;
typedef __attribute__((ext_vector_type(8))) float v8f;

// CDNA5 fp32 WMMA: D(16x16,f32) = A(16x4,f32) * B(4x16,f32) + C
__device__ __forceinline__ v8f wmma4(v2f a, v2f b, v8f c) {
  return __builtin_amdgcn_wmma_f32_16x16x4_f32(
      /*neg_a=*/false, a, /*neg_b=*/false, b,
      /*c_mod=*/(short)0, c, /*reuse_a=*/false, /*reuse_b=*/false);
}

__device__ __forceinline__ float sigf(float x) { return 1.0f / (1.0f + __expf(-x)); }

// Async global->LDS copy (CDNA5).  LDS byte address = low 32 bits of the
// generic shared pointer (flat aperture rule, ISA 10.2).
__device__ __forceinline__ void async_load_b128(void* lds_dst, const void* gsrc) {
  unsigned ldsa = (unsigned)(unsigned long long)lds_dst;
  asm volatile("global_load_async_to_lds_b128 %0, %1, off"
               :: "v"(ldsa), "v"(gsrc) : "memory");
}
__device__ __forceinline__ void wait_async0() {
  asm volatile("s_wait_asynccnt 0x0" ::: "memory");
}

// ---------------------------------------------------------------------------
// Utility fills
// ---------------------------------------------------------------------------
__global__ void fill_f32(float* p, float v, size_t n) {
  size_t i = (size_t)blockIdx.x * blockDim.x + threadIdx.x;
  if (i < n) p[i] = v;
}
__global__ void fill_i32(int* p, int v, size_t n) {
  size_t i = (size_t)blockIdx.x * blockDim.x + threadIdx.x;
  if (i < n) p[i] = v;
}

// ---------------------------------------------------------------------------
// Degree / norm
// ---------------------------------------------------------------------------
__global__ void edge_degree(const int* __restrict__ ei, float* __restrict__ deg) {
  int e = blockIdx.x * blockDim.x + threadIdx.x;
  if (e >= N_EDGES) return;
  atomicAdd(&deg[ei[N_EDGES + e]], 1.0f);
}
__global__ void make_dinv(const float* __restrict__ deg, float* __restrict__ dinv) {
  int i = blockIdx.x * blockDim.x + threadIdx.x;
  if (i >= N_NODES) return;
  dinv[i] = rsqrtf(fmaxf(deg[i], 1.0f));
}

// ---------------------------------------------------------------------------
// WMMA fp32 GEMM, compile-time K and strides: C = A[M,K]*B[K,N] (+bias)
// One wave32 per 16x16 tile; fully unrolled; branch-free masked K tail;
// dual accumulators to break the WMMA D->C dependency chain.
// ---------------------------------------------------------------------------
template <int K, int LDA, int LDB, int LDC, bool HAS_BIAS>
__global__ void wmma_gemm_t(const float* __restrict__ A, const float* __restrict__ B,
                            const float* __restrict__ bias, float* __restrict__ C) {
  const int lane = threadIdx.x;              // 0..31
  const int tm = blockIdx.x, tn = blockIdx.y;
  const int mrow = tm * 16 + (lane & 15);
  const int ncol = tn * 16 + (lane & 15);
  const int koff = (lane < 16) ? 0 : 2;
  constexpr int KF = K & ~3;

  const float* __restrict__ Arow = A + (size_t)mrow * LDA;
  const float* __restrict__ Bcol = B + ncol;

  v8f acc0 = {0.f, 0.f, 0.f, 0.f, 0.f, 0.f, 0.f, 0.f};
  v8f acc1 = {0.f, 0.f, 0.f, 0.f, 0.f, 0.f, 0.f, 0.f};
#pragma unroll
  for (int k0 = 0; k0 < KF; k0 += 4) {
    const int ka = k0 + koff;
    v2f a, b;
    a.x = Arow[ka];
    a.y = Arow[ka + 1];
    b.x = Bcol[(size_t)ka * LDB];
    b.y = Bcol[(size_t)(ka + 1) * LDB];
    if ((k0 >> 2) & 1) acc1 = wmma4(a, b, acc1);
    else               acc0 = wmma4(a, b, acc0);
  }
  if constexpr ((K & 3) != 0) {              // masked tail (K=54)
    const int ka = KF + koff;
    const float m0 = (ka     < K) ? 1.f : 0.f;
    const float m1 = (ka + 1 < K) ? 1.f : 0.f;
    const int c0 = (ka     < K) ? ka     : (K - 1);
    const int c1 = (ka + 1 < K) ? ka + 1 : (K - 1);
    v2f a, b;
    a.x = Arow[c0] * m0;
    a.y = Arow[c1] * m1;
    b.x = Bcol[(size_t)c0 * LDB] * m0;
    b.y = Bcol[(size_t)c1 * LDB] * m1;
    acc0 = wmma4(a, b, acc0);
  }

  const int mbase = tm * 16 + ((lane < 16) ? 0 : 8);
  const float bv = HAS_BIAS ? bias[ncol] : 0.f;
#pragma unroll
  for (int r = 0; r < 8; ++r)
    C[(size_t)(mbase + r) * LDC + ncol] = acc0[r] + acc1[r] + bv;
}

// ---------------------------------------------------------------------------
// GCN aggregation
// ---------------------------------------------------------------------------
__global__ void self_loop_init(const float4* __restrict__ xw, const float* __restrict__ dinv,
                               float4* __restrict__ agg) {
  size_t t = (size_t)blockIdx.x * blockDim.x + threadIdx.x;   // N*16/4
  if (t >= (size_t)N_NODES * H_GCN / 4) return;
  float di = dinv[t >> 2];
  float s = di * di;
  float4 v = xw[t];
  v.x *= s; v.y *= s; v.z *= s; v.w *= s;
  agg[t] = v;
}

__global__ void edge_scatter(const int* __restrict__ ei, const float* __restrict__ xw,
                             const float* __restrict__ dinv, float* __restrict__ agg) {
  int e = blockIdx.x * blockDim.x + threadIdx.x;
  if (e >= N_EDGES) return;
  int s = ei[e];
  int d = ei[N_EDGES + e];
  float norm = dinv[s] * dinv[d];
  const float4* __restrict__ xr = (const float4*)&xw[(size_t)s * H_GCN];
  float4 x0 = xr[0], x1 = xr[1], x2 = xr[2], x3 = xr[3];
  float* ar = &agg[(size_t)d * H_GCN];
  atomicAdd(&ar[0],  x0.x * norm); atomicAdd(&ar[1],  x0.y * norm);
  atomicAdd(&ar[2],  x0.z * norm); atomicAdd(&ar[3],  x0.w * norm);
  atomicAdd(&ar[4],  x1.x * norm); atomicAdd(&ar[5],  x1.y * norm);
  atomicAdd(&ar[6],  x1.z * norm); atomicAdd(&ar[7],  x1.w * norm);
  atomicAdd(&ar[8],  x2.x * norm); atomicAdd(&ar[9],  x2.y * norm);
  atomicAdd(&ar[10], x2.z * norm); atomicAdd(&ar[11], x2.w * norm);
  atomicAdd(&ar[12], x3.x * norm); atomicAdd(&ar[13], x3.y * norm);
  atomicAdd(&ar[14], x3.z * norm); atomicAdd(&ar[15], x3.w * norm);
}

__global__ void bias_relu(const float4* __restrict__ agg, const float* __restrict__ b,
                          float4* __restrict__ h) {
  size_t t = (size_t)blockIdx.x * blockDim.x + threadIdx.x;   // N*16/4
  if (t >= (size_t)N_NODES * H_GCN / 4) return;
  int fb = (int)((t & 3) << 2);
  float4 v = agg[t];
  v.x = fmaxf(v.x + b[fb],     0.f);
  v.y = fmaxf(v.y + b[fb + 1], 0.f);
  v.z = fmaxf(v.z + b[fb + 2], 0.f);
  v.w = fmaxf(v.w + b[fb + 3], 0.f);
  h[t] = v;
}

// ---------------------------------------------------------------------------
// Ragged -> padded packing
// ---------------------------------------------------------------------------
__global__ void count_nodes(const int* __restrict__ batch, int* __restrict__ counts) {
  int i = blockIdx.x * blockDim.x + threadIdx.x;
  if (i >= N_NODES) return;
  atomicAdd(&counts[batch[i]], 1);
}

__global__ void scan_starts(const int* __restrict__ counts, int* __restrict__ starts) {
  __shared__ int tmp[NUM_GRAPHS];
  int i = threadIdx.x;
  tmp[i] = counts[i];
  __syncthreads();
  for (int off = 1; off < NUM_GRAPHS; off <<= 1) {
    int v = (i >= off) ? tmp[i - off] : 0;
    __syncthreads();
    tmp[i] += v;
    __syncthreads();
  }
  starts[i] = tmp[i] - counts[i];  // exclusive prefix
}

__global__ void pack_padded(const float4* __restrict__ h, const int* __restrict__ batch,
                            const int* __restrict__ starts, float4* __restrict__ padded) {
  size_t t = (size_t)blockIdx.x * blockDim.x + threadIdx.x;   // N*16/4
  if (t >= (size_t)N_NODES * H_GCN / 4) return;
  int i = (int)(t >> 2);
  int f4 = (int)(t & 3);
  int b = batch[i];
  int pos = i - starts[b];
  if (pos < SEQ_T)
    padded[(((size_t)b * SEQ_T + pos) * H_GCN >> 2) + f4] = h[t];
}

// ---------------------------------------------------------------------------
// BiLSTM layer, compile-time input width D.
// Block = 256 threads (8 waves); one wave per 16-wide gate tile (N=128).
// x_t is double-buffered in LDS and filled with global_load_async_to_lds_b128
// one step ahead (ASYNCcnt pipeline); h/c state and fused weights live in LDS.
// Per step: G[16,128] = [x_t | h](16 x Kc) @ Wcat^T(Kc x 128) on fp32 WMMA.
// ---------------------------------------------------------------------------
template <int D>
__global__ void lstm_layer_t(const float* __restrict__ in_seq,
                             const float* __restrict__ wih, const float* __restrict__ whh,
                             const float* __restrict__ bih, const float* __restrict__ bhh,
                             float* __restrict__ out_seq) {
  constexpr int Kc = D + H_LSTM;
  constexpr int CHUNKS = 4 * D;            // 16-byte chunks per x_t tile (16 rows)
  __shared__ float Wcat[4 * H_LSTM][Kc];   // fused [Wih | Whh], row-major
  __shared__ float bias[4 * H_LSTM];
  __shared__ float xbuf[2][16][D];         // double-buffered x_t
  __shared__ float hst[16][H_LSTM];        // hidden state
  __shared__ float cst[16][H_LSTM];        // cell state
  __shared__ float gbuf[16][4 * H_LSTM];   // gate pre-activations

  const int tid = threadIdx.x;             // 0..255
  const int lane = tid & 31;
  const int wv = tid >> 5;                 // wave id = gate tile nn (0..7)
  const int gtile = blockIdx.x;            // 0..31
  const int dir = blockIdx.y;              // 0 fwd, 1 bwd
  const int b0 = gtile * 16;

  for (int idx = tid; idx < 4 * H_LSTM * Kc; idx += 256) {
    int g = idx / Kc, k = idx % Kc;
    Wcat[g][k] = (k < D) ? wih[((size_t)dir * 4 * H_LSTM + g) * D + k]
                         : whh[((size_t)dir * 4 * H_LSTM + g) * H_LSTM + (k - D)];
  }
  if (tid < 4 * H_LSTM) bias[tid] = bih[dir * 4 * H_LSTM + tid] + bhh[dir * 4 * H_LSTM + tid];
  for (int idx = tid; idx < 16 * H_LSTM; idx += 256) {
    int m = idx / H_LSTM, hu = idx % H_LSTM;
    cst[m][hu] = 0.f;
    hst[m][hu] = 0.f;
  }

  // prologue: async-stage x_{t0} into buffer 0
  {
    const int t0 = dir ? (SEQ_T - 1) : 0;
    if (tid < CHUNKS) {
      int m = tid / (D / 4), kq = (tid % (D / 4)) * 4;
      async_load_b128(&xbuf[0][m][kq],
                      &in_seq[((size_t)(b0 + m) * SEQ_T + t0) * D + kq]);
    }
    wait_async0();
  }
  __syncthreads();

  const int mrow = lane & 15;
  const int koff = (lane < 16) ? 0 : 2;
  const int mb = (lane < 16) ? 0 : 8;
  const int ncol = wv * 16 + (lane & 15);  // this wave's gate column
  int cur = 0;

  for (int step = 0; step < SEQ_T; ++step) {
    const int t = dir ? (SEQ_T - 1 - step) : step;

    // issue async stage of x_{t+1} into the other buffer (overlaps compute)
    if (step + 1 < SEQ_T) {
      const int tn = dir ? (SEQ_T - 2 - step) : (step + 1);
      if (tid < CHUNKS) {
        int m = tid / (D / 4), kq = (tid % (D / 4)) * 4;
        async_load_b128(&xbuf[cur ^ 1][m][kq],
                        &in_seq[((size_t)(b0 + m) * SEQ_T + tn) * D + kq]);
      }
    }

    // G tile nn=wv : [16 x Kc] @ [Kc x 16] via fully unrolled fp32 WMMA
    {
      const float* __restrict__ ax = &xbuf[cur][mrow][0];
      const float* __restrict__ ah = &hst[mrow][0];
      const float* __restrict__ wrow = &Wcat[ncol][0];
      v8f acc0 = {0.f, 0.f, 0.f, 0.f, 0.f, 0.f, 0.f, 0.f};
      v8f acc1 = {0.f, 0.f, 0.f, 0.f, 0.f, 0.f, 0.f, 0.f};
#pragma unroll
      for (int k0 = 0; k0 < Kc; k0 += 4) {
        const int ka = k0 + koff;
        v2f a, b;
        if (k0 < D) { a.x = ax[ka];     a.y = ax[ka + 1]; }
        else        { a.x = ah[ka - D]; a.y = ah[ka - D + 1]; }
        b.x = wrow[ka]; b.y = wrow[ka + 1];
        if ((k0 >> 2) & 1) acc1 = wmma4(a, b, acc1);
        else               acc0 = wmma4(a, b, acc0);
      }
      const float bv = bias[ncol];
#pragma unroll
      for (int r = 0; r < 8; ++r) gbuf[mb + r][ncol] = acc0[r] + acc1[r] + bv;
    }
    __syncthreads();

    // gate nonlinearity + state update (PyTorch order i,f,g,o): 2 per thread
#pragma unroll
    for (int j = 0; j < 2; ++j) {
      int idx = tid * 2 + j;               // 16 graphs x 32 hidden = 512
      int m = idx >> 5, hu = idx & 31;
      float ig = gbuf[m][hu];
      float fg = gbuf[m][H_LSTM + hu];
      float gg = gbuf[m][2 * H_LSTM + hu];
      float og = gbuf[m][3 * H_LSTM + hu];
      float c = sigf(fg) * cst[m][hu] + sigf(ig) * tanhf(gg);
      float h = sigf(og) * tanhf(c);
      cst[m][hu] = c;
      hst[m][hu] = h;
      out_seq[((size_t)(b0 + m) * SEQ_T + t) * (2 * H_LSTM) + dir * H_LSTM + hu] = h;
    }

    wait_async0();        // next x tile landed in LDS (issuing waves)
    __syncthreads();      // visibility of hst + xbuf[cur^1] to all waves
    cur ^= 1;
  }
}

// ---------------------------------------------------------------------------
// Mean pool over time (float4 per thread)
// ---------------------------------------------------------------------------
__global__ void mean_pool(const float4* __restrict__ seq, float4* __restrict__ pooled) {
  int t = blockIdx.x * blockDim.x + threadIdx.x;   // 512 * 16
  if (t >= NUM_GRAPHS * (2 * H_LSTM / 4)) return;
  int b = t / (2 * H_LSTM / 4), j4 = t % (2 * H_LSTM / 4);
  float4 s = {0.f, 0.f, 0.f, 0.f};
  const float4* p = &seq[(size_t)b * SEQ_T * (2 * H_LSTM / 4) + j4];
  for (int k = 0; k < SEQ_T; ++k) {
    float4 v = p[(size_t)k * (2 * H_LSTM / 4)];
    s.x += v.x; s.y += v.y; s.z += v.z; s.w += v.w;
  }
  const float r = 1.0f / SEQ_T;
  s.x *= r; s.y *= r; s.z *= r; s.w *= r;
  pooled[t] = s;
}

// ---------------------------------------------------------------------------
// Host-side launch
// ---------------------------------------------------------------------------
static inline size_t align256(size_t x) { return (x + 255) & ~(size_t)255; }

extern "C" void kernel_launch(void* const* d_in, const int* in_sizes, int n_in,
                              void* d_out, int out_size, void* d_ws, size_t ws_size,
                              hipStream_t stream) {
  (void)in_sizes; (void)n_in; (void)out_size; (void)ws_size;

  const float* x        = (const float*)d_in[0];
  const int*   ei       = (const int*)  d_in[1];
  const int*   batch    = (const int*)  d_in[2];
  const float* gcn_w1   = (const float*)d_in[3];
  const float* gcn_b1   = (const float*)d_in[4];
  const float* gcn_w2   = (const float*)d_in[5];
  const float* gcn_b2   = (const float*)d_in[6];
  const float* l0_wih   = (const float*)d_in[7];
  const float* l0_whh   = (const float*)d_in[8];
  const float* l0_bih   = (const float*)d_in[9];
  const float* l0_bhh   = (const float*)d_in[10];
  const float* l1_wih   = (const float*)d_in[11];
  const float* l1_whh   = (const float*)d_in[12];
  const float* l1_bih   = (const float*)d_in[13];
  const float* l1_bhh   = (const float*)d_in[14];
  const float* fc_w     = (const float*)d_in[15];
  const float* fc_b     = (const float*)d_in[16];
  float* out = (float*)d_out;

  char* ws = (char*)d_ws;
  size_t off = 0;
  float* deg    = (float*)(ws + off); off = align256(off + (size_t)N_NODES * 4);
  float* dinv   = (float*)(ws + off); off = align256(off + (size_t)N_NODES * 4);
  float* xw     = (float*)(ws + off); off = align256(off + (size_t)N_NODES * H_GCN * 4);
  float* agg    = (float*)(ws + off); off = align256(off + (size_t)N_NODES * H_GCN * 4);
  float* hbuf   = (float*)(ws + off); off = align256(off + (size_t)N_NODES * H_GCN * 4);
  int*   counts = (int*)  (ws + off); off = align256(off + (size_t)NUM_GRAPHS * 4);
  int*   starts = (int*)  (ws + off); off = align256(off + (size_t)NUM_GRAPHS * 4);
  float* padded = (float*)(ws + off); off = align256(off + (size_t)NUM_GRAPHS * SEQ_T * H_GCN * 4);
  float* seq0   = (float*)(ws + off); off = align256(off + (size_t)NUM_GRAPHS * SEQ_T * 2 * H_LSTM * 4);
  float* seq1   = (float*)(ws + off); off = align256(off + (size_t)NUM_GRAPHS * SEQ_T * 2 * H_LSTM * 4);
  float* pooled = (float*)(ws + off); off = align256(off + (size_t)NUM_GRAPHS * 2 * H_LSTM * 4);

  const int B256 = 256;
  const size_t NF4 = (size_t)N_NODES * H_GCN / 4;

  // ---- GCN normalization ----
  fill_f32<<<(N_NODES + B256 - 1) / B256, B256, 0, stream>>>(deg, 1.0f, N_NODES); // self loops
  edge_degree<<<(N_EDGES + B256 - 1) / B256, B256, 0, stream>>>(ei, deg);
  make_dinv<<<(N_NODES + B256 - 1) / B256, B256, 0, stream>>>(deg, dinv);

  // ---- GCN layer 1 ----
  wmma_gemm_t<F_IN, F_IN, H_GCN, H_GCN, false>
      <<<dim3(N_NODES / 16, H_GCN / 16), 32, 0, stream>>>(x, gcn_w1, nullptr, xw);
  self_loop_init<<<(NF4 + B256 - 1) / B256, B256, 0, stream>>>(
      (const float4*)xw, dinv, (float4*)agg);
  edge_scatter<<<(N_EDGES + B256 - 1) / B256, B256, 0, stream>>>(ei, xw, dinv, agg);
  bias_relu<<<(NF4 + B256 - 1) / B256, B256, 0, stream>>>(
      (const float4*)agg, gcn_b1, (float4*)hbuf);

  // ---- GCN layer 2 ----
  wmma_gemm_t<H_GCN, H_GCN, H_GCN, H_GCN, false>
      <<<dim3(N_NODES / 16, H_GCN / 16), 32, 0, stream>>>(hbuf, gcn_w2, nullptr, xw);
  self_loop_init<<<(NF4 + B256 - 1) / B256, B256, 0, stream>>>(
      (const float4*)xw, dinv, (float4*)agg);
  edge_scatter<<<(N_EDGES + B256 - 1) / B256, B256, 0, stream>>>(ei, xw, dinv, agg);
  bias_relu<<<(NF4 + B256 - 1) / B256, B256, 0, stream>>>(
      (const float4*)agg, gcn_b2, (float4*)hbuf);

  // ---- pack ragged -> padded [512,1024,16] ----
  fill_i32<<<(NUM_GRAPHS + B256 - 1) / B256, B256, 0, stream>>>(counts, 0, NUM_GRAPHS);
  count_nodes<<<(N_NODES + B256 - 1) / B256, B256, 0, stream>>>(batch, counts);
  scan_starts<<<1, NUM_GRAPHS, 0, stream>>>(counts, starts);
  {
    size_t pn = (size_t)NUM_GRAPHS * SEQ_T * H_GCN;
    fill_f32<<<(pn + B256 - 1) / B256, B256, 0, stream>>>(padded, 0.f, pn);
  }
  pack_padded<<<(NF4 + B256 - 1) / B256, B256, 0, stream>>>(
      (const float4*)hbuf, batch, starts, (float4*)padded);

  // ---- BiLSTM layer 0: [512,1024,16] -> [512,1024,64] ----
  lstm_layer_t<H_GCN><<<dim3(NUM_GRAPHS / 16, 2), 256, 0, stream>>>(
      padded, l0_wih, l0_whh, l0_bih, l0_bhh, seq0);

  // ---- BiLSTM layer 1: [512,1024,64] -> [512,1024,64] ----
  lstm_layer_t<2 * H_LSTM><<<dim3(NUM_GRAPHS / 16, 2), 256, 0, stream>>>(
      seq0, l1_wih, l1_whh, l1_bih, l1_bhh, seq1);

  // ---- mean pool + FC ----
  mean_pool<<<(NUM_GRAPHS * (2 * H_LSTM / 4) + B256 - 1) / B256, B256, 0, stream>>>(
      (const float4*)seq1, (float4*)pooled);
  wmma_gemm_t<2 * H_LSTM, 2 * H_LSTM, N_CLASSES, N_CLASSES, true>
      <<<dim3(NUM_GRAPHS / 16, N_CLASSES / 16), 32, 0, stream>>>(pooled, fc_w, fc_b, out);
}